// SingleLayerDecoder_8710193677086
// MI455X (gfx1250) — compile-verified
//
#include <hip/hip_runtime.h>
#include <hip/hip_bf16.h>
#include <math.h>

// Problem constants from the reference: B=16, L=256, H=256, D=256
#define BB   16
#define LL   256
#define HH   256
#define DD   256
#define M_ROWS (BB * LL)        // 4096
#define BIG_NEG 1e8f
#define PAD  260                // LDS row pitch: 16B-aligned rows, 260%64=4 -> no bank conflicts

typedef __attribute__((ext_vector_type(2))) float v2f;
typedef __attribute__((ext_vector_type(8))) float v8f;

// 8 MB of scratch for dotL / dotR; fully overwritten by kernel 1 every call
// before kernel 2 reads it -> deterministic, no reliance on ws_size.
__device__ float g_dotL[M_ROWS * DD];
__device__ float g_dotR[M_ROWS * DD];

#if __has_builtin(__builtin_amdgcn_tanhf)
static __device__ __forceinline__ float fast_tanh(float x) { return __builtin_amdgcn_tanhf(x); }
#elif __has_builtin(__builtin_amdgcn_tanh_f32)
static __device__ __forceinline__ float fast_tanh(float x) { return __builtin_amdgcn_tanh_f32(x); }
#else
static __device__ __forceinline__ float fast_tanh(float x) { return tanhf(x); }
#endif

static __device__ __forceinline__ void wait_asynccnt0() {
#if __has_builtin(__builtin_amdgcn_s_wait_asynccnt)
    __builtin_amdgcn_s_wait_asynccnt(0);
#else
    asm volatile("s_wait_asynccnt 0x0" ::: "memory");
#endif
}

// ---------------------------------------------------------------------------
// Kernel 2 (defined first so the disasm snippet shows the tanh hot loop):
// logits[b,i,j] = sum_d tanh(dotL[b,i,d] + dotR[b,j,d]) * u[d] + bias
// then diagonal mask, sigmoid, hash-RNG Bernoulli sample, Bernoulli entropy.
// Block = 256 threads handles one (b, i-tile) and loops over the 16 j-tiles.
// sR tiles staged with GLOBAL_LOAD_ASYNC_TO_LDS_B128 (ASYNCcnt-tracked DMA).
// ---------------------------------------------------------------------------
__global__ __launch_bounds__(256) void score_kernel(const float* __restrict__ u,
                                                    const float* __restrict__ l_bias,
                                                    float* __restrict__ out) {
    __shared__ __align__(16) float sL[16][PAD];
    __shared__ __align__(16) float sR[16][PAD];
    __shared__ __align__(16) float su[DD];

    const int tid = threadIdx.x;
    const int it  = blockIdx.x;        // i tile: 0..15
    const int b   = blockIdx.y;        // batch:  0..15
    const float bias = l_bias[0];

    // stage dotL tile [16 x 256] (float4 chunks) and u
    const size_t rowL0 = (size_t)(b * LL + it * 16) * DD;
    #pragma unroll
    for (int k = 0; k < 4; ++k) {
        const int c   = tid + k * 256;       // chunk id 0..1023
        const int r   = c >> 6;              // 64 float4-chunks per row
        const int col = (c & 63) << 2;
        *(float4*)&sL[r][col] = *(const float4*)&g_dotL[rowL0 + (size_t)r * DD + col];
    }
    su[tid] = u[tid];

    // LDS byte offset of sR: low 32 bits of the generic shared pointer
    const unsigned sR_base = (unsigned)(uintptr_t)&sR[0][0];

    const int i = tid >> 4;            // 0..15
    const int j = tid & 15;            // 0..15
    const int gi = it * 16 + i;

    float* __restrict__ out_samp = out;
    float* __restrict__ out_mask = out + (size_t)BB * LL * LL;
    float* __restrict__ out_ent  = out + 2 * (size_t)BB * LL * LL;

    for (int jt = 0; jt < 16; ++jt) {
        __syncthreads();               // all waves done reading previous sR

        // async DMA: global (GVS: SGPR base + 32-bit VGPR offset) -> LDS
        const float* gbase = &g_dotR[(size_t)(b * LL + jt * 16) * DD];  // uniform
        #pragma unroll
        for (int k = 0; k < 4; ++k) {
            const int c   = tid + k * 256;
            const int r   = c >> 6;
            const int col = (c & 63) << 2;
            const unsigned lds  = sR_base + (unsigned)(r * PAD + col) * 4u;
            const unsigned voff = (unsigned)(r * DD + col) * 4u;
            asm volatile("global_load_async_to_lds_b128 %0, %1, %2"
                         :
                         : "v"(lds), "v"(voff), "s"(gbase)
                         : "memory");
        }
        wait_asynccnt0();
        __syncthreads();               // every wave's DMA has landed

        // 268M-tanh hot loop: one (i,j) output per thread, reduce over d
        float acc = 0.0f;
        #pragma unroll 2
        for (int d = 0; d < DD; d += 4) {
            const float4 lv = *(const float4*)&sL[i][d];
            const float4 rv = *(const float4*)&sR[j][d];
            const float4 uv = *(const float4*)&su[d];
            acc += fast_tanh(lv.x + rv.x) * uv.x;
            acc += fast_tanh(lv.y + rv.y) * uv.y;
            acc += fast_tanh(lv.z + rv.z) * uv.z;
            acc += fast_tanh(lv.w + rv.w) * uv.w;
        }

        const int gj = jt * 16 + j;
        float x = acc + bias - ((gi == gj) ? BIG_NEG : 0.0f);

        const float p  = 1.0f / (1.0f + __expf(-x));                       // sigmoid
        const float sp = fmaxf(x, 0.0f) + log1pf(__expf(-fabsf(x)));       // softplus
        const float ent = sp - x * p;

        // deterministic PCG-style hash RNG (stand-in for JAX threefry bits)
        unsigned int h = (unsigned int)(((b * LL) + gi) * LL + gj);
        h = h * 747796405u + 2891336453u;
        h = ((h >> ((h >> 28) + 4u)) ^ h) * 277803737u;
        h = (h >> 22) ^ h;
        const float u01 = (float)(h >> 8) * (1.0f / 16777216.0f);
        const float samp = (u01 < p) ? 1.0f : 0.0f;

        const size_t o = ((size_t)(b * LL) + gi) * LL + gj;
        out_samp[o] = samp;
        out_mask[o] = x;
        out_ent[o]  = ent;
    }
}

// ---------------------------------------------------------------------------
// Kernel 1: fused dual GEMM  dotL = X @ W_l,  dotR = X @ W_r
// One wave32 computes one 16x16 fp32 output tile via V_WMMA_F32_16X16X4_F32.
// Tile id space: [2 matrices] x [M/16 = 256] x [N/16 = 16] = 8192 wave-tiles.
// ---------------------------------------------------------------------------
__global__ __launch_bounds__(256) void dual_gemm_wmma(const float* __restrict__ X,
                                                      const float* __restrict__ Wl,
                                                      const float* __restrict__ Wr) {
    const int lane = threadIdx.x & 31;
    const int wave = threadIdx.x >> 5;
    const int tile = blockIdx.x * 8 + wave;        // 0 .. 8191

    const int mat = tile >> 12;                    // 0 = L, 1 = R (4096 tiles each)
    const int t   = tile & 4095;
    const int mt  = t >> 4;                        // 0..255  (M tile)
    const int nt  = t & 15;                        // 0..15   (N tile)

    const float* __restrict__ W   = mat ? Wr : Wl;
    float* __restrict__       Dst = mat ? g_dotR : g_dotL;

    const int half = lane >> 4;                    // 0: K pair {0,1}, 1: K pair {2,3}
    const int l15  = lane & 15;

    const int arow = mt * 16 + l15;                // A row this lane feeds
    const int bcol = nt * 16 + l15;                // B col this lane feeds

    const float* __restrict__ Xrow = X + (size_t)arow * HH;

    v8f acc = {};
    #pragma unroll 4
    for (int k = 0; k < HH; k += 4) {
        const int ka = k + half * 2;
        // A fragment: 16x4 fp32. lanes 0-15 hold K=0,1 ; lanes 16-31 hold K=2,3
        v2f a;
        a.x = Xrow[ka];
        a.y = Xrow[ka + 1];
        // B fragment: 4x16 fp32, rows striped across lanes within a VGPR.
        // lanes 0-15: K rows {0,1} ; lanes 16-31: K rows {2,3}
        v2f bf;
        bf.x = W[(size_t)ka * DD + bcol];
        bf.y = W[(size_t)(ka + 1) * DD + bcol];
        acc = __builtin_amdgcn_wmma_f32_16x16x4_f32(
            /*neg_a=*/false, a, /*neg_b=*/false, bf,
            /*c_mod=*/(short)0, acc, /*reuse_a=*/false, /*reuse_b=*/false);
    }

    // C/D layout: VGPR r -> lanes 0-15: M=r ; lanes 16-31: M=r+8 ; N = lane&15
    #pragma unroll
    for (int r = 0; r < 8; ++r) {
        const int m = half ? (r + 8) : r;
        Dst[(size_t)(mt * 16 + m) * DD + nt * 16 + l15] = acc[r];
    }
}

extern "C" void kernel_launch(void* const* d_in, const int* in_sizes, int n_in,
                              void* d_out, int out_size, void* d_ws, size_t ws_size,
                              hipStream_t stream) {
    (void)in_sizes; (void)n_in; (void)out_size; (void)d_ws; (void)ws_size;
    const float* X      = (const float*)d_in[0];   // [B, L, H]
    const float* Wl     = (const float*)d_in[1];   // [H, D]
    const float* Wr     = (const float*)d_in[2];   // [H, D]
    const float* u      = (const float*)d_in[3];   // [D]
    const float* l_bias = (const float*)d_in[4];   // [1]
    float* out = (float*)d_out;                    // samples | masked | entropy

    // 8192 wave-tiles / 8 waves per block = 1024 blocks
    dual_gemm_wmma<<<1024, 256, 0, stream>>>(X, Wl, Wr);
    score_kernel<<<dim3(LL / 16, BB), 256, 0, stream>>>(u, l_bias, out);
}